// localGNN_46222438039625
// MI455X (gfx1250) — compile-verified
//
#include <hip/hip_runtime.h>

// ---------------------------------------------------------------------------
// localGNN on gfx1250: per-subgraph fused GIN using V_WMMA_F32_16X16X4_F32
//   G=256, MAXN=64, NPS=16, H=128, S=16384 subgraphs, N=262144, E=1048576
// v3: weights staged in LDS pre-swizzled into B-operand layout so every B
//     fetch is one aligned ds_load_b64 (no register moves); 8 waves/block.
// ---------------------------------------------------------------------------

typedef __attribute__((ext_vector_type(2))) float v2f;
typedef __attribute__((ext_vector_type(4))) float v4f;
typedef __attribute__((ext_vector_type(8))) float v8f;

#define HDIM   128
#define S_TOT  16384
#define WAVES  8           // waves (=subgraphs) per block
#define NTHR   (WAVES * 32)
#define RSA    132         // activation row stride (floats): A b64 reads conflict-free
#define PW2    144         // weight pair-row pitch (float2): 2*144 % 64 == 32
                           //   -> half-waves hit complementary bank sets

__device__ __forceinline__ float gelu_exact(float x) {
    // torch GELU default: 0.5*x*(1+erf(x/sqrt(2)))
    return 0.5f * x * (1.0f + erff(x * 0.70710678118654752f));
}

// Cooperative copy of one 128x128 weight matrix into LDS, swizzled into
// B-operand layout: wl2[(k>>1)*PW2 + col] = { W[k][col], W[k+1][col] }.
__device__ __forceinline__ void coop_load_W(v2f* wl2, const float* __restrict__ W) {
    const int t = threadIdx.x;
    #pragma unroll
    for (int i = 0; i < (64 * 32) / NTHR; ++i) {   // 64 row-pairs x 32 col-quads
        const int idx = t + i * NTHR;
        const int p = idx >> 5;                    // row pair 0..63
        const int g = (idx & 31) << 2;             // col base 0,4,...,124
        v4f r0 = *(const v4f*)(W + (2 * p)     * HDIM + g);
        v4f r1 = *(const v4f*)(W + (2 * p + 1) * HDIM + g);
        v4f lo, hi;
        lo[0] = r0[0]; lo[1] = r1[0]; lo[2] = r0[1]; lo[3] = r1[1];
        hi[0] = r0[2]; hi[1] = r1[2]; hi[2] = r0[3]; hi[3] = r1[3];
        *(v4f*)(&wl2[p * PW2 + g])     = lo;       // cols g, g+1
        *(v4f*)(&wl2[p * PW2 + g + 2]) = hi;       // cols g+2, g+3
    }
}

// buf[16][128] = gelu( buf[16][128] @ W + bias ), in place; W pre-swizzled in LDS.
// All 8 output tiles accumulate in registers; kt outer -> A ds_load_b64 shared
// by 8 independent wmma issues; each B operand is one aligned ds_load_b64.
__device__ __forceinline__ void dense_layer(
    float* __restrict__ buf, const v2f* wl2,
    const float* __restrict__ bias, int lm, int half)
{
    v8f acc[8];
    #pragma unroll
    for (int nt = 0; nt < 8; ++nt) {
        const float b = bias[nt * 16 + lm];
        #pragma unroll
        for (int r = 0; r < 8; ++r) acc[nt][r] = b;        // bias broadcast C-init
    }
    #pragma unroll 2
    for (int kt = 0; kt < 32; ++kt) {
        const int k0 = kt * 4 + half * 2;                  // K-chunk of 4
        const int pr = kt * 2 + half;                      // pair-row = k0>>1
        v2f a;
        a[0] = buf[lm * RSA + k0];
        a[1] = buf[lm * RSA + k0 + 1];
        #pragma unroll
        for (int nt = 0; nt < 8; ++nt) {
            v2f bb = wl2[pr * PW2 + nt * 16 + lm];         // {W[k0][col], W[k0+1][col]}
            acc[nt] = __builtin_amdgcn_wmma_f32_16x16x4_f32(
                false, a, false, bb, (short)0, acc[nt], false, false);
        }
    }
    #pragma unroll
    for (int nt = 0; nt < 8; ++nt) {
        #pragma unroll
        for (int r = 0; r < 8; ++r)
            buf[(r + 8 * half) * RSA + nt * 16 + lm] = gelu_exact(acc[nt][r]);
    }
}

__global__ void zero_f32_kernel(float* __restrict__ p, int n) {
    int i = blockIdx.x * blockDim.x + threadIdx.x;
    if (i < n) p[i] = 0.0f;
}

// Acnt[sub][i][j] = #edges with dst-slot i, src-slot j inside subgraph `sub`
__global__ void count_edges_kernel(const int* __restrict__ ei,
                                   float* __restrict__ Acnt, int E) {
    int e = blockIdx.x * blockDim.x + threadIdx.x;
    if (e < E) {
        int s = ei[e];          // src node id
        int d = ei[E + e];      // dst node id (same subgraph as src)
        int sub = d >> 4;
        atomicAdd(&Acnt[(sub << 8) + ((d & 15) << 4) + (s & 15)], 1.0f);
    }
}

__global__ __launch_bounds__(NTHR)
void gnn_fused_kernel(
    const float* __restrict__ Acnt,
    const float* __restrict__ W1a, const float* __restrict__ b1a,
    const float* __restrict__ W1b, const float* __restrict__ b1b,
    const float* __restrict__ W2a, const float* __restrict__ b2a,
    const float* __restrict__ W2b, const float* __restrict__ b2b,
    const float* __restrict__ Wf0, const float* __restrict__ bf0,
    const float* __restrict__ Wf1, const float* __restrict__ bf1,
    float* __restrict__ out)
{
    __shared__ float act[WAVES * 16 * RSA];   // one 16x128 buffer per wave
    __shared__ v2f   wl2[64 * PW2];           // current weights, B-operand layout

    const int wave = threadIdx.x >> 5;
    const int lane = threadIdx.x & 31;
    const int lm   = lane & 15;
    const int half = lane >> 4;
    const int sub  = blockIdx.x * WAVES + wave;

    float* buf = &act[wave * 16 * RSA];
    const float* Ag = Acnt + (size_t)sub * 256;

    // stage W1b while computing layer-1a (no dependency)
    coop_load_W(wl2, W1b);

    // ---- in-degree of each of the 16 nodes: row-sum of count matrix --------
    float deg = 0.0f;
    if (lane < 16) {
        #pragma unroll
        for (int j = 0; j < 16; ++j) deg += Ag[lane * 16 + j];
    }
    // ---- layer 1a: h1a[i][c] = gelu((1+deg_i) * W1a[c] + b1a[c]) -> buf ----
    float w1c[4], b1c[4];
    #pragma unroll
    for (int cc = 0; cc < 4; ++cc) {
        w1c[cc] = W1a[lane + cc * 32];
        b1c[cc] = b1a[lane + cc * 32];
    }
    for (int i = 0; i < 16; ++i) {
        float d = 1.0f + __shfl(deg, i, 32);
        #pragma unroll
        for (int cc = 0; cc < 4; ++cc)
            buf[i * RSA + lane + cc * 32] = gelu_exact(d * w1c[cc] + b1c[cc]);
    }
    __syncthreads();

    // ---- h1 = gelu(h1a @ W1b + b1b) ----------------------------------------
    dense_layer(buf, wl2, b1b, lm, half);
    __syncthreads();                       // everyone done reading W1b

    // stage W2a while doing the (weight-free) aggregation
    coop_load_W(wl2, W2a);

    // ---- layer-2 aggregation: t = (A + I) @ h1, in place -------------------
    {
        v8f acc[8];
        #pragma unroll
        for (int nt = 0; nt < 8; ++nt) {
            const int col = nt * 16 + lm;
            #pragma unroll
            for (int r = 0; r < 8; ++r)
                acc[nt][r] = buf[(r + 8 * half) * RSA + col];   // +I via C-init
        }
        #pragma unroll
        for (int kt = 0; kt < 4; ++kt) {
            const int k0 = kt * 4 + half * 2;
            v2f a;
            a[0] = Ag[lm * 16 + k0];
            a[1] = Ag[lm * 16 + k0 + 1];
            #pragma unroll
            for (int nt = 0; nt < 8; ++nt) {
                const int col = nt * 16 + lm;
                v2f bb;
                bb[0] = buf[k0 * RSA + col];
                bb[1] = buf[(k0 + 1) * RSA + col];
                acc[nt] = __builtin_amdgcn_wmma_f32_16x16x4_f32(
                    false, a, false, bb, (short)0, acc[nt], false, false);
            }
        }
        #pragma unroll
        for (int nt = 0; nt < 8; ++nt) {
            #pragma unroll
            for (int r = 0; r < 8; ++r)
                buf[(r + 8 * half) * RSA + nt * 16 + lm] = acc[nt][r];
        }
    }
    __syncthreads();

    dense_layer(buf, wl2, b2a, lm, half);   // gelu(t @ W2a + b2a)
    __syncthreads();
    coop_load_W(wl2, W2b);
    __syncthreads();
    dense_layer(buf, wl2, b2b, lm, half);   // gelu(. @ W2b + b2b)
    __syncthreads();
    coop_load_W(wl2, Wf0);
    __syncthreads();
    dense_layer(buf, wl2, bf0, lm, half);   // gelu(. @ Wf0 + bf0)

    // ---- pool over 16 nodes, dot with Wf1, + bf1 ---------------------------
    float part = 0.0f;
    #pragma unroll
    for (int cc = 0; cc < 4; ++cc) {
        const int c = lane + cc * 32;
        const float wf = Wf1[c];
        float s = 0.0f;
        #pragma unroll
        for (int i = 0; i < 16; ++i) s += buf[i * RSA + c];
        part += s * wf;
    }
    #pragma unroll
    for (int off = 16; off > 0; off >>= 1)
        part += __shfl_down(part, off, 32);
    if (lane == 0) out[sub] = part + bf1[0];   // out layout [G][MAXN] == flat sub id
}

extern "C" void kernel_launch(void* const* d_in, const int* in_sizes, int n_in,
                              void* d_out, int out_size, void* d_ws, size_t ws_size,
                              hipStream_t stream)
{
    const int*   ei  = (const int*)d_in[0];       // edge_index [2][E]
    const float* W1a = (const float*)d_in[4];
    const float* b1a = (const float*)d_in[5];
    const float* W1b = (const float*)d_in[6];
    const float* b1b = (const float*)d_in[7];
    const float* W2a = (const float*)d_in[8];
    const float* b2a = (const float*)d_in[9];
    const float* W2b = (const float*)d_in[10];
    const float* b2b = (const float*)d_in[11];
    const float* Wf0 = (const float*)d_in[12];
    const float* bf0 = (const float*)d_in[13];
    const float* Wf1 = (const float*)d_in[14];
    const float* bf1 = (const float*)d_in[15];
    float* out  = (float*)d_out;
    float* Acnt = (float*)d_ws;                   // S * 256 floats = 16 MB

    const int E  = in_sizes[0] / 2;
    const int nA = S_TOT * 256;

    zero_f32_kernel<<<(nA + 255) / 256, 256, 0, stream>>>(Acnt, nA);
    count_edges_kernel<<<(E + 255) / 256, 256, 0, stream>>>(ei, Acnt, E);
    gnn_fused_kernel<<<S_TOT / WAVES, NTHR, 0, stream>>>(
        Acnt, W1a, b1a, W1b, b1b, W2a, b2a, W2b, b2b, Wf0, bf0, Wf1, bf1, out);
}